// RegLoss_34608846471207
// MI455X (gfx1250) — compile-verified
//
#include <hip/hip_runtime.h>
#include <hip/hip_bf16.h>

// Problem sizes fixed by the reference setup_inputs():
#define VV 4
#define BB 8
#define NN 2048
#define DD 128
#define VIEWS_ELEMS (VV * BB * NN * DD)   // 8388608
#define GRAPH_ELEMS (BB * NN * NN)        // 33554432
#define COS_EPS 1e-3f
#define INV_EPS 1e3f                      // 1/COS_EPS
#define ALPHA 0.1f
#define BETA  0.1f

// Super-tile decomposition for the Gram kernel: 32x32 per wave (2x2 WMMA tiles)
#define NST 64                            // NN/32 super-tiles per dim
#define PT  (NST * (NST + 1) / 2)         // 2080 triangular super-tiles per batch

// Grid sizing
#define NB_R 1024   // recon partial blocks
#define NB_G 2048   // graph partial blocks
#define NB_C 512    // cosine/WMMA partial blocks
#define BLK  256    // 8 wave32 per block

typedef __attribute__((ext_vector_type(2))) float v2f;
typedef __attribute__((ext_vector_type(4))) float v4f;   // clang vector: OK for nontemporal builtins
typedef __attribute__((ext_vector_type(8))) float v8f;

__device__ inline float waveReduce32(float v) {
#pragma unroll
    for (int m = 16; m; m >>= 1) v += __shfl_xor(v, m, 32);
    return v;
}

// ---------------------------------------------------------------------------
// Kernel 1: per-row reciprocal L2 norms of latent_embeddings [BB*NN x DD].
// One wave per row: 32 lanes x 4 floats = 128 elements.
// Storing 1/norm turns the 34M hot-path divides into multiplies:
//   1/max(n_i*n_j, eps) == min(rn_i*rn_j, 1/eps)   (exact, incl. n==0 -> inf)
// ---------------------------------------------------------------------------
__global__ void k_rnorms(const float* __restrict__ E, float* __restrict__ rn) {
    const int gwave = (blockIdx.x * blockDim.x + threadIdx.x) >> 5;
    const int lane  = threadIdx.x & 31;
    if (gwave >= BB * NN) return;                 // wave-uniform
    const v4f* row = (const v4f*)(E + (size_t)gwave * DD);
    v4f v = row[lane];
    float s = v.x * v.x + v.y * v.y + v.z * v.z + v.w * v.w;
    s = waveReduce32(s);
    if (lane == 0) rn[gwave] = 1.0f / sqrtf(s);
}

// ---------------------------------------------------------------------------
// Kernel 2: cosine loss via fp32 WMMA (V_WMMA_F32_16X16X4_F32).
// Each wave computes one 32x32 super-tile of the Gram matrix as 2x2 register-
// blocked 16x16 WMMA tiles (4 WMMAs per 4 b64 fragment loads -> half the L2
// traffic of unblocked). Symmetry of |G|/denom lets us enumerate only
// sti <= stj and weight off-diagonal super-tiles x2 -> half again.
//
// Fragment layouts per CDNA5 ISA 7.12.2 (f32):
//   A 16x4 : lanes 0-15 -> M = lane, VGPR0/1 = K 0/1; lanes 16-31 -> K 2/3.
//   B 4x16 : mirrored with N across lanes.
//   C/D    : VGPR v -> row v + 8*(lane>=16), col = lane&15.
// ---------------------------------------------------------------------------
__global__ void k_cos(const float* __restrict__ E, const float* __restrict__ rn,
                      float* __restrict__ part) {
    const int lane  = threadIdx.x & 31;
    const int wib   = threadIdx.x >> 5;
    const int wpb   = blockDim.x >> 5;
    const int gwave = blockIdx.x * wpb + wib;
    const int nwav  = gridDim.x * wpb;
    const int half  = lane >> 4;     // 0/1 -> K offset 0 / +2
    const int l15   = lane & 15;
    const int nSuper = BB * PT;

    float wsum = 0.f;
    for (int t = gwave; t < nSuper; t += nwav) {       // wave-uniform loop
        const int bb = t / PT;
        const int p  = t - bb * PT;
        // triangular decode: p -> (sti <= stj)
        int r = (int)((sqrtf(8.f * (float)p + 1.f) - 1.f) * 0.5f);
        while ((r + 1) * (r + 2) / 2 <= p) ++r;
        while (r * (r + 1) / 2 > p) --r;
        const int stj = r;
        const int sti = p - r * (r + 1) / 2;
        const float wgt = (sti == stj) ? 1.f : 2.f;

        const float* Eb  = E + (size_t)bb * NN * DD;
        const float* pa0 = Eb + (size_t)(sti * 32 + l15) * DD + half * 2;
        const float* pa1 = pa0 + 16 * DD;
        const float* pb0 = Eb + (size_t)(stj * 32 + l15) * DD + half * 2;
        const float* pb1 = pb0 + 16 * DD;

        v8f a00 = {0.f,0.f,0.f,0.f,0.f,0.f,0.f,0.f};
        v8f a01 = a00, a10 = a00, a11 = a00;
#pragma unroll 4
        for (int k = 0; k < DD; k += 4) {
            v2f av0 = *(const v2f*)(pa0 + k);
            v2f av1 = *(const v2f*)(pa1 + k);
            v2f bv0 = *(const v2f*)(pb0 + k);
            v2f bv1 = *(const v2f*)(pb1 + k);
            a00 = __builtin_amdgcn_wmma_f32_16x16x4_f32(false, av0, false, bv0, (short)0, a00, false, false);
            a01 = __builtin_amdgcn_wmma_f32_16x16x4_f32(false, av0, false, bv1, (short)0, a01, false, false);
            a10 = __builtin_amdgcn_wmma_f32_16x16x4_f32(false, av1, false, bv0, (short)0, a10, false, false);
            a11 = __builtin_amdgcn_wmma_f32_16x16x4_f32(false, av1, false, bv1, (short)0, a11, false, false);
        }

        const float* rnb = rn + bb * NN;
        const float rc0 = rnb[stj * 32 + l15];
        const float rc1 = rnb[stj * 32 + 16 + l15];
        float tsum = 0.f;
#pragma unroll
        for (int v = 0; v < 8; ++v) {
            const int m = v + half * 8;
            const float rr0 = rnb[sti * 32 + m];
            const float rr1 = rnb[sti * 32 + 16 + m];
            tsum += fabsf(a00[v]) * fminf(rr0 * rc0, INV_EPS);
            tsum += fabsf(a01[v]) * fminf(rr0 * rc1, INV_EPS);
            tsum += fabsf(a10[v]) * fminf(rr1 * rc0, INV_EPS);
            tsum += fabsf(a11[v]) * fminf(rr1 * rc1, INV_EPS);
        }
        wsum += wgt * tsum;
    }

    wsum = waveReduce32(wsum);
    __shared__ float sh[BLK / 32];
    if (lane == 0) sh[wib] = wsum;
    __syncthreads();
    if (threadIdx.x == 0) {
        float s = 0.f;
#pragma unroll
        for (int i = 0; i < BLK / 32; ++i) s += sh[i];
        part[blockIdx.x] = s;
    }
}

// ---------------------------------------------------------------------------
// Kernel 3: sum of squared diff over views / views_recon. Single-use stream:
// nontemporal loads keep it from evicting the L2-resident latent tensor.
// ---------------------------------------------------------------------------
__global__ void k_recon(const v4f* __restrict__ a, const v4f* __restrict__ b,
                        float* __restrict__ part) {
    const int n4 = VIEWS_ELEMS / 4;
    float s = 0.f;
    for (int i = blockIdx.x * blockDim.x + threadIdx.x; i < n4;
         i += gridDim.x * blockDim.x) {
        v4f x = __builtin_nontemporal_load(&a[i]);
        v4f y = __builtin_nontemporal_load(&b[i]);
        float dx = x.x - y.x, dy = x.y - y.y, dz = x.z - y.z, dw = x.w - y.w;
        s += dx * dx + dy * dy + dz * dz + dw * dw;
    }
    s = waveReduce32(s);
    __shared__ float sh[BLK / 32];
    const int lane = threadIdx.x & 31, w = threadIdx.x >> 5;
    if (lane == 0) sh[w] = s;
    __syncthreads();
    if (threadIdx.x == 0) {
        float t = 0.f;
#pragma unroll
        for (int i = 0; i < BLK / 32; ++i) t += sh[i];
        part[blockIdx.x] = t;
    }
}

// ---------------------------------------------------------------------------
// Kernel 4: single fused pass over input_graph / recon_graph (268 MB, the HBM
// roofline term): sq = sum (in-rec)^2, ab = sum |in - I|. Nontemporal.
// ---------------------------------------------------------------------------
__global__ void k_graph(const v4f* __restrict__ g_in, const v4f* __restrict__ g_rec,
                        float* __restrict__ part) {
    const unsigned n4 = GRAPH_ELEMS / 4;
    float sq = 0.f, ab = 0.f;
    for (unsigned i = blockIdx.x * blockDim.x + threadIdx.x; i < n4;
         i += gridDim.x * blockDim.x) {
        v4f x = __builtin_nontemporal_load(&g_in[i]);
        v4f y = __builtin_nontemporal_load(&g_rec[i]);
        float dx = x.x - y.x, dy = x.y - y.y, dz = x.z - y.z, dw = x.w - y.w;
        sq += dx * dx + dy * dy + dz * dz + dw * dw;
        const unsigned f  = i * 4u;               // flat float index
        const unsigned j0 = f & (NN - 1);         // column of first component
        const unsigned ii = (f >> 11) & (NN - 1); // row (NN = 2^11)
        ab += fabsf(x.x - ((ii == j0 + 0u) ? 1.f : 0.f));
        ab += fabsf(x.y - ((ii == j0 + 1u) ? 1.f : 0.f));
        ab += fabsf(x.z - ((ii == j0 + 2u) ? 1.f : 0.f));
        ab += fabsf(x.w - ((ii == j0 + 3u) ? 1.f : 0.f));
    }
    sq = waveReduce32(sq);
    ab = waveReduce32(ab);
    __shared__ float shq[BLK / 32], sha[BLK / 32];
    const int lane = threadIdx.x & 31, w = threadIdx.x >> 5;
    if (lane == 0) { shq[w] = sq; sha[w] = ab; }
    __syncthreads();
    if (threadIdx.x == 0) {
        float tq = 0.f, ta = 0.f;
#pragma unroll
        for (int i = 0; i < BLK / 32; ++i) { tq += shq[i]; ta += sha[i]; }
        part[2 * blockIdx.x + 0] = tq;
        part[2 * blockIdx.x + 1] = ta;
    }
}

// ---------------------------------------------------------------------------
// Kernel 5: combine partials, apply scalar post-ops, emit the 5 outputs.
// ---------------------------------------------------------------------------
__global__ void k_final(const float* __restrict__ partR,
                        const float* __restrict__ partG,
                        const float* __restrict__ partC,
                        float* __restrict__ out) {
    __shared__ float sh[BLK / 32];
    const int lane = threadIdx.x & 31, w = threadIdx.x >> 5;

    float sR = 0.f, sGq = 0.f, sGa = 0.f, sC = 0.f;
    for (int i = threadIdx.x; i < NB_R; i += blockDim.x) sR += partR[i];
    for (int i = threadIdx.x; i < NB_G; i += blockDim.x) {
        sGq += partG[2 * i + 0];
        sGa += partG[2 * i + 1];
    }
    for (int i = threadIdx.x; i < NB_C; i += blockDim.x) sC += partC[i];

    float vals[4] = {sR, sGq, sGa, sC};
    float red[4];
#pragma unroll
    for (int q = 0; q < 4; ++q) {
        float v = waveReduce32(vals[q]);
        if (lane == 0) sh[w] = v;
        __syncthreads();
        if (threadIdx.x == 0) {
            float t = 0.f;
#pragma unroll
            for (int i = 0; i < BLK / 32; ++i) t += sh[i];
            red[q] = t;
        }
        __syncthreads();
    }

    if (threadIdx.x == 0) {
        const float recon_loss = red[0] / (float)VIEWS_ELEMS;
        const float rgl        = sqrtf(red[1]) / (float)BB;
        const float gl         = red[2] / (float)BB;
        // sum(|g|/denom) - b*n reproduces the "- eye" term exactly
        const float cos_loss   = (red[3] - (float)(BB * NN)) /
                                 ((float)BB * (float)NN * (float)(NN - 1));
        const float loss = recon_loss + rgl + ALPHA * cos_loss + BETA * gl;
        out[0] = loss;
        out[1] = recon_loss;
        out[2] = rgl;
        out[3] = cos_loss;
        out[4] = gl;
    }
}

// ---------------------------------------------------------------------------
extern "C" void kernel_launch(void* const* d_in, const int* in_sizes, int n_in,
                              void* d_out, int out_size, void* d_ws, size_t ws_size,
                              hipStream_t stream) {
    const float* views       = (const float*)d_in[0];
    const float* views_recon = (const float*)d_in[1];
    const float* latent      = (const float*)d_in[2];
    const float* g_in        = (const float*)d_in[3];
    const float* g_rec       = (const float*)d_in[4];
    float* out = (float*)d_out;

    // Workspace layout (floats):
    float* ws     = (float*)d_ws;
    float* rnorms = ws;                        // 16384
    float* partR  = rnorms + BB * NN;          // NB_R
    float* partG  = partR + NB_R;              // 2 * NB_G
    float* partC  = partG + 2 * NB_G;          // NB_C
    (void)in_sizes; (void)n_in; (void)out_size; (void)ws_size;

    // Order: latent-consuming kernels first so the 8 MB latent tensor stays
    // L2-resident between them; the 268 MB graph stream runs last (and uses
    // nontemporal loads anyway).
    k_rnorms<<<(BB * NN) / (BLK / 32), BLK, 0, stream>>>(latent, rnorms);
    k_cos<<<NB_C, BLK, 0, stream>>>(latent, rnorms, partC);
    k_recon<<<NB_R, BLK, 0, stream>>>((const v4f*)views,
                                      (const v4f*)views_recon, partR);
    k_graph<<<NB_G, BLK, 0, stream>>>((const v4f*)g_in,
                                      (const v4f*)g_rec, partG);
    k_final<<<1, BLK, 0, stream>>>(partR, partG, partC, out);
}